// MultiscaleAttention_62199716380938
// MI455X (gfx1250) — compile-verified
//
#include <hip/hip_runtime.h>
#include <math.h>

typedef __attribute__((ext_vector_type(2))) float v2f;
typedef __attribute__((ext_vector_type(8))) float v8f;

// Problem constants
#define BB     4
#define NTOK   6273          // 1 + 8*28*28
#define EMB    768
#define NHD    8
#define HD     96
#define N2     1569          // 1 + 8*14*14
#define SPQ    1568          // 8*14*14
#define M_QKV  (BB*NTOK)     // 25092
#define M_PROJ (BB*N2)       // 6276

#define KC     32            // GEMM K-chunk
#define ASTR   36            // padded LDS stride (bank-conflict-free, 16B-aligned)
#define KVSTR  100           // padded K/V tile stride
#define PSTR   20            // padded P tile stride

// ---------------------------------------------------------------------------
// fp32 WMMA GEMM: C[M,Nn] = A[M,K]*W[Nn,K]^T + bias[Nn]
// 8 waves (256 thr), 64x32 block tile, K-chunk 32, double-buffered async
// global->LDS staging (GLOBAL_LOAD_ASYNC_TO_LDS_B128 + s_wait_asynccnt).
// Requires K % 32 == 0, Nn % 32 == 0.
// ---------------------------------------------------------------------------
__global__ __launch_bounds__(256) void gemm_wmma_f32(
    const float* __restrict__ A, const float* __restrict__ W,
    const float* __restrict__ bias, float* __restrict__ C,
    int M, int Nn, int K)
{
    __shared__ __align__(16) float Abuf[2][64 * ASTR];
    __shared__ __align__(16) float Wbuf[2][32 * ASTR];

    const int tid  = threadIdx.x;
    const int lane = tid & 31;
    const int w    = tid >> 5;
    const int l15  = lane & 15;
    const int koff = (lane < 16) ? 0 : 2;
    const int mw   = w >> 1, nw = w & 1;
    const int m0b  = blockIdx.x * 64;
    const int n0b  = blockIdx.y * 32;

    // Async copy segment assignment (float4 granularity)
    // A tile: 64 rows x 32 -> 512 segs; this wave: segs w*64+lane, +32
    const int aseg0 = w * 64 + lane, aseg1 = aseg0 + 32;
    const int ar0 = aseg0 >> 3, ac0 = (aseg0 & 7) * 4;
    const int ar1 = aseg1 >> 3, ac1 = (aseg1 & 7) * 4;
    int g0 = m0b + ar0; if (g0 >= M) g0 = M - 1;   // clamp: garbage rows unused
    int g1 = m0b + ar1; if (g1 >= M) g1 = M - 1;
    const float* gA0 = A + (size_t)g0 * K + ac0;
    const float* gA1 = A + (size_t)g1 * K + ac1;
    // W tile: 32 rows x 32 -> 256 segs; this wave: seg w*32+lane
    const int wseg = w * 32 + lane;
    const int wr = wseg >> 3, wc = (wseg & 7) * 4;
    const float* gW = W + (size_t)(n0b + wr) * K + wc;

    auto issue = [&](int kc, int buf) {
        uint32_t lA0 = (uint32_t)(uintptr_t)&Abuf[buf][ar0 * ASTR + ac0];
        uint32_t lA1 = (uint32_t)(uintptr_t)&Abuf[buf][ar1 * ASTR + ac1];
        uint32_t lW  = (uint32_t)(uintptr_t)&Wbuf[buf][wr * ASTR + wc];
        const float* p0 = gA0 + kc;
        const float* p1 = gA1 + kc;
        const float* p2 = gW + kc;
        asm volatile("global_load_async_to_lds_b128 %0, %1, off"
                     :: "v"(lA0), "v"(p0) : "memory");
        asm volatile("global_load_async_to_lds_b128 %0, %1, off"
                     :: "v"(lA1), "v"(p1) : "memory");
        asm volatile("global_load_async_to_lds_b128 %0, %1, off"
                     :: "v"(lW), "v"(p2) : "memory");
    };

    const int nchunks = K / KC;     // 24 for K=768
    issue(0, 0);

    v8f acc = {0.f,0.f,0.f,0.f,0.f,0.f,0.f,0.f};
    const int arow = mw * 16 + l15;
    const int wrow = nw * 16 + l15;

    for (int c = 0; c < nchunks; ++c) {
        if (c + 1 < nchunks) {
            issue((c + 1) * KC, (c + 1) & 1);
            asm volatile("s_wait_asynccnt 0x3" ::: "memory");
        } else {
            asm volatile("s_wait_asynccnt 0x0" ::: "memory");
        }
        __syncthreads();                        // chunk c visible to all waves
        const float* As = &Abuf[c & 1][arow * ASTR];
        const float* Ws = &Wbuf[c & 1][wrow * ASTR];
#pragma unroll
        for (int k = 0; k < KC; k += 4) {
            v2f a = *(const v2f*)(As + k + koff);
            v2f b = *(const v2f*)(Ws + k + koff);
            acc = __builtin_amdgcn_wmma_f32_16x16x4_f32(false, a, false, b,
                                                        (short)0, acc, false, false);
        }
        __syncthreads();                        // done reading before overwrite
    }

    const int col = n0b + nw * 16 + l15;
    const float bv = bias[col];
    const int m0 = m0b + mw * 16;
#pragma unroll
    for (int r = 0; r < 8; ++r) {
        const int row = m0 + r + ((lane < 16) ? 0 : 8);
        if (row < M) C[(size_t)row * Nn + col] = acc[r] + bv;
    }
}

// ---------------------------------------------------------------------------
// Pool (depthwise 3x3x3 conv, stride (1,2,2), pad 1) + LayerNorm over HD.
// grid = (N2, B*NH, 3[part: q,k,v]), block = 96 (one thread per channel d)
// ---------------------------------------------------------------------------
__global__ __launch_bounds__(96) void pool_ln_kernel(
    const float* __restrict__ qkv,
    const float* __restrict__ kq, const float* __restrict__ kk, const float* __restrict__ kv,
    const float* __restrict__ gq, const float* __restrict__ bq,
    const float* __restrict__ gk, const float* __restrict__ bk,
    const float* __restrict__ gv, const float* __restrict__ bv,
    float* __restrict__ qp, float* __restrict__ kp, float* __restrict__ vp)
{
    const int n2   = blockIdx.x;
    const int bh   = blockIdx.y;
    const int part = blockIdx.z;
    const int d    = threadIdx.x;
    const int b    = bh >> 3, h = bh & 7;

    const float* kern = (part == 0) ? kq : (part == 1) ? kk : kv;
    const float* g    = (part == 0) ? gq : (part == 1) ? gk : gv;
    const float* be   = (part == 0) ? bq : (part == 1) ? bk : bv;
    float*       dst  = (part == 0) ? qp : (part == 1) ? kp : vp;

    const size_t cbase = (size_t)part * EMB + (size_t)h * HD + d;

    float val;
    if (n2 == 0) {
        val = qkv[(size_t)(b * NTOK) * 2304 + cbase];
    } else {
        const int s  = n2 - 1;
        const int t2 = s / 196, rem = s % 196, y2 = rem / 14, x2 = rem % 14;
        float acc = 0.f;
#pragma unroll
        for (int dt = 0; dt < 3; ++dt) {
            const int t = t2 + dt - 1;
            if (t < 0 || t >= 8) continue;
#pragma unroll
            for (int dy = 0; dy < 3; ++dy) {
                const int y = 2 * y2 + dy - 1;
                if (y < 0 || y >= 28) continue;
#pragma unroll
                for (int dx = 0; dx < 3; ++dx) {
                    const int x = 2 * x2 + dx - 1;
                    if (x < 0 || x >= 28) continue;
                    const int n = 1 + t * 784 + y * 28 + x;
                    acc += kern[d * 27 + dt * 9 + dy * 3 + dx] *
                           qkv[(size_t)(b * NTOK + n) * 2304 + cbase];
                }
            }
        }
        val = acc;
    }

    __shared__ float sh[HD];
    sh[d] = val;
    __syncthreads();
    float mean = 0.f;
    for (int i = 0; i < HD; ++i) mean += sh[i];
    mean *= (1.f / HD);
    float var = 0.f;
    for (int i = 0; i < HD; ++i) { float t = sh[i] - mean; var += t * t; }
    var *= (1.f / HD);
    const float o = (val - mean) * rsqrtf(var + 1e-5f) * g[d] + be[d];
    dst[(size_t)(bh * N2 + n2) * HD + d] = o;
}

// ---------------------------------------------------------------------------
// Decomposed rel-pos bias per pooled (non-cls) query:
// relb[bh][qi][0..7]=rel_t, [8..21]=rel_h, [22..35]=rel_w
// ---------------------------------------------------------------------------
__global__ __launch_bounds__(64) void relbias_kernel(
    const float* __restrict__ qpool,
    const float* __restrict__ rph, const float* __restrict__ rpw,
    const float* __restrict__ rpt, float* __restrict__ relb)
{
    const int qi = blockIdx.x;
    const int bh = blockIdx.y;
    __shared__ float qv[HD];
    const float* qptr = qpool + (size_t)(bh * N2 + 1 + qi) * HD;
    for (int i = threadIdx.x; i < HD; i += 64) qv[i] = qptr[i];
    __syncthreads();

    const int t = threadIdx.x;
    if (t < 36) {
        const int qt = qi / 196, rem = qi % 196, qy = rem / 14, qx = rem % 14;
        const float* table; int idx;
        if (t < 8)       { table = rpt; idx = qt - t + 7; }
        else if (t < 22) { table = rph; idx = qy - (t - 8) + 13; }
        else             { table = rpw; idx = qx - (t - 22) + 13; }
        const float* r = table + (size_t)idx * HD;
        float acc = 0.f;
        for (int c = 0; c < HD; ++c) acc += qv[c] * r[c];
        relb[(size_t)(bh * SPQ + qi) * 36 + t] = acc;
    }
}

// ---------------------------------------------------------------------------
// Fused flash attention with WMMA (fp32 16x16x4), online softmax, rel-pos
// bias gather, residual, head-interleaved output. K/V tiles double-buffered
// via async global->LDS copies.
// grid = (25, 32), block = 128 (4 waves, 16 queries each)
// ---------------------------------------------------------------------------
__global__ __launch_bounds__(128) void attn_flash_wmma(
    const float* __restrict__ qp, const float* __restrict__ kp,
    const float* __restrict__ vp, const float* __restrict__ relb,
    float* __restrict__ aout)
{
    __shared__ __align__(16) float Kbuf[2][16 * KVSTR];
    __shared__ __align__(16) float Vbuf[2][16 * KVSTR];
    __shared__ __align__(16) float Pbuf[4][16 * PSTR];
    __shared__ __align__(16) float Rq[4][16 * 36];

    const int tid  = threadIdx.x;
    const int lane = tid & 31;
    const int w    = tid >> 5;
    const int l15  = lane & 15;
    const int koff = (lane < 16) ? 0 : 2;
    const int half = (lane < 16) ? 0 : 8;

    const int bh = blockIdx.y;
    const int b  = bh >> 3, h = bh & 7;
    const int q0 = blockIdx.x * 64 + w * 16;
    const float scaler = 0.1020620726159658f;   // 1/sqrt(96)

    // async K/V tile staging: 16 keys x 96 ch = 384 float4 segs per tensor;
    // this wave: segs w*96 + i*32 + lane, i=0..2  (3 b128 per tensor per tile)
    auto issueKV = [&](int k0, int buf) {
#pragma unroll
        for (int i = 0; i < 3; ++i) {
            const int seg = w * 96 + i * 32 + lane;
            const int key = seg / 24;
            const int c4  = (seg % 24) * 4;
            int kg = k0 + key; if (kg >= N2) kg = N2 - 1;  // clamp; masked later
            const float* gk = kp + (size_t)(bh * N2 + kg) * HD + c4;
            const float* gv = vp + (size_t)(bh * N2 + kg) * HD + c4;
            uint32_t lk = (uint32_t)(uintptr_t)&Kbuf[buf][key * KVSTR + c4];
            uint32_t lv = (uint32_t)(uintptr_t)&Vbuf[buf][key * KVSTR + c4];
            asm volatile("global_load_async_to_lds_b128 %0, %1, off"
                         :: "v"(lk), "v"(gk) : "memory");
            asm volatile("global_load_async_to_lds_b128 %0, %1, off"
                         :: "v"(lv), "v"(gv) : "memory");
        }
    };

    issueKV(0, 0);

    // Q A-fragments (scaled), kept resident
    int mrow = q0 + l15; if (mrow >= N2) mrow = N2 - 1;
    const float* qbase = qp + (size_t)(bh * N2 + mrow) * HD + koff;
    v2f qa[24];
#pragma unroll
    for (int s = 0; s < 24; ++s) {
        v2f t = *(const v2f*)(qbase + s * 4);
        qa[s].x = t.x * scaler;
        qa[s].y = t.y * scaler;
    }

    // rel-pos rows for this wave's 16 queries
    for (int i = lane; i < 16 * 36; i += 32) {
        const int r  = i / 36;
        const int qg = q0 + r;
        float vv = 0.f;
        if (qg >= 1 && qg < N2)
            vv = relb[(size_t)(bh * SPQ + (qg - 1)) * 36 + (i % 36)];
        Rq[w][i] = vv;
    }

    v8f O[6];
#pragma unroll
    for (int j = 0; j < 6; ++j) O[j] = (v8f){0.f,0.f,0.f,0.f,0.f,0.f,0.f,0.f};
    float mstat[8], lstat[8];
#pragma unroll
    for (int r = 0; r < 8; ++r) { mstat[r] = -1e30f; lstat[r] = 0.f; }

    for (int ktile = 0; ktile < 99; ++ktile) {
        const int k0 = ktile * 16;
        if (ktile + 1 < 99) {
            issueKV((ktile + 1) * 16, (ktile + 1) & 1);
            asm volatile("s_wait_asynccnt 0x6" ::: "memory");
        } else {
            asm volatile("s_wait_asynccnt 0x0" ::: "memory");
        }
        __syncthreads();                    // tile ktile visible block-wide
        const float* Kt = Kbuf[ktile & 1];
        const float* Vt = Vbuf[ktile & 1];

        // S = (Q*scale) @ K^T
        v8f S = {0.f,0.f,0.f,0.f,0.f,0.f,0.f,0.f};
#pragma unroll
        for (int s = 0; s < 24; ++s) {
            v2f bf = *(const v2f*)(Kt + l15 * KVSTR + s * 4 + koff);
            S = __builtin_amdgcn_wmma_f32_16x16x4_f32(false, qa[s], false, bf,
                                                      (short)0, S, false, false);
        }

        // bias + mask + online softmax per row
#pragma unroll
        for (int r = 0; r < 8; ++r) {
            const int m  = r + half;
            const int qg = q0 + m;
            const int kg = k0 + l15;
            float bb;
            if (kg >= N2) bb = -1e30f;
            else if (qg >= 1 && qg < N2 && kg >= 1) {
                const int ki = kg - 1;
                const int kt_ = ki / 196, krm = ki % 196;
                const int ky = krm / 14, kx = krm % 14;
                bb = Rq[w][m * 36 + kt_] + Rq[w][m * 36 + 8 + ky] +
                     Rq[w][m * 36 + 22 + kx];
            } else bb = 0.f;

            float sv = S[r] + bb;
            float mx = sv;
#pragma unroll
            for (int msk = 1; msk <= 8; msk <<= 1)
                mx = fmaxf(mx, __shfl_xor(mx, msk, 32));
            const float mnew = fmaxf(mstat[r], mx);
            const float sc   = __expf(mstat[r] - mnew);
            const float p    = __expf(sv - mnew);
            float ps = p;
#pragma unroll
            for (int msk = 1; msk <= 8; msk <<= 1)
                ps += __shfl_xor(ps, msk, 32);
            lstat[r] = lstat[r] * sc + ps;
            mstat[r] = mnew;
#pragma unroll
            for (int j = 0; j < 6; ++j) O[j][r] *= sc;
            Pbuf[w][m * PSTR + l15] = p;
        }

        // O += P @ V  (A = P via per-wave LDS round trip, B = Vt)
#pragma unroll
        for (int s2 = 0; s2 < 4; ++s2) {
            const int kk = s2 * 4 + koff;
            v2f pa = *(const v2f*)(&Pbuf[w][l15 * PSTR + kk]);
#pragma unroll
            for (int j = 0; j < 6; ++j) {
                v2f vb;
                vb.x = Vt[kk * KVSTR + j * 16 + l15];
                vb.y = Vt[(kk + 1) * KVSTR + j * 16 + l15];
                O[j] = __builtin_amdgcn_wmma_f32_16x16x4_f32(false, pa, false, vb,
                                                             (short)0, O[j],
                                                             false, false);
            }
        }
        __syncthreads();                    // all reads done before overwrite
    }

    // epilogue: normalize, residual (+q_pool for non-cls), head-interleave
#pragma unroll
    for (int r = 0; r < 8; ++r) {
        const int qg = q0 + r + half;
        if (qg >= N2) continue;
        const float inv = 1.f / lstat[r];
#pragma unroll
        for (int j = 0; j < 6; ++j) {
            const int d = j * 16 + l15;
            float o = O[j][r] * inv;
            if (qg >= 1) o += qp[(size_t)(bh * N2 + qg) * HD + d];
            aout[(size_t)(b * N2 + qg) * EMB + h * HD + d] = o;
        }
    }
}

// ---------------------------------------------------------------------------
extern "C" void kernel_launch(void* const* d_in, const int* in_sizes, int n_in,
                              void* d_out, int out_size, void* d_ws, size_t ws_size,
                              hipStream_t stream)
{
    (void)in_sizes; (void)n_in; (void)out_size; (void)ws_size;
    const float* x      = (const float*)d_in[0];
    const float* qkv_w  = (const float*)d_in[1];
    const float* qkv_b  = (const float*)d_in[2];
    const float* poolq  = (const float*)d_in[3];
    const float* poolk  = (const float*)d_in[4];
    const float* poolv  = (const float*)d_in[5];
    const float* lnq_g  = (const float*)d_in[6];
    const float* lnq_b  = (const float*)d_in[7];
    const float* lnk_g  = (const float*)d_in[8];
    const float* lnk_b  = (const float*)d_in[9];
    const float* lnv_g  = (const float*)d_in[10];
    const float* lnv_b  = (const float*)d_in[11];
    const float* rph    = (const float*)d_in[12];
    const float* rpw    = (const float*)d_in[13];
    const float* rpt    = (const float*)d_in[14];
    const float* proj_w = (const float*)d_in[15];
    const float* proj_b = (const float*)d_in[16];
    float* out = (float*)d_out;

    float* ws = (float*)d_ws;
    const size_t pool_elems = (size_t)BB * NHD * N2 * HD;
    float* qkv  = ws;
    float* qp   = qkv + (size_t)M_QKV * 2304;
    float* kp   = qp + pool_elems;
    float* vp   = kp + pool_elems;
    float* rb   = vp + pool_elems;
    float* ao   = rb + (size_t)BB * NHD * SPQ * 36;

    gemm_wmma_f32<<<dim3((M_QKV + 63) / 64, 2304 / 32), 256, 0, stream>>>(
        x, qkv_w, qkv_b, qkv, M_QKV, 2304, EMB);

    pool_ln_kernel<<<dim3(N2, BB * NHD, 3), 96, 0, stream>>>(
        qkv, poolq, poolk, poolv, lnq_g, lnq_b, lnk_g, lnk_b, lnv_g, lnv_b,
        qp, kp, vp);

    relbias_kernel<<<dim3(SPQ, BB * NHD), 64, 0, stream>>>(qp, rph, rpw, rpt, rb);

    attn_flash_wmma<<<dim3((N2 + 63) / 64, BB * NHD), 128, 0, stream>>>(
        qp, kp, vp, rb, ao);

    gemm_wmma_f32<<<dim3((M_PROJ + 63) / 64, EMB / 32), 256, 0, stream>>>(
        ao, proj_w, proj_b, out, M_PROJ, EMB, EMB);
}